// NR_GraphAttention_14559939133751
// MI455X (gfx1250) — compile-verified
//
#include <hip/hip_runtime.h>

#define DIM    128
#define DIMO   384   // DIM*(DEPTH+1)
#define NPROXY 64

typedef __attribute__((ext_vector_type(2))) float v2f;
typedef __attribute__((ext_vector_type(8))) float v8f;

__device__ __forceinline__ float atomicMaxF(float* addr, float v) {
  // sound for mixed signs: int-max for >=0, uint-min for <0
  if (v >= 0.0f)
    return __int_as_float(atomicMax((int*)addr, __float_as_int(v)));
  else
    return __uint_as_float(atomicMin((unsigned int*)addr, __float_as_uint(v)));
}

__device__ __forceinline__ float waveRedSum(float x) {
#pragma unroll
  for (int off = 16; off > 0; off >>= 1) x += __shfl_xor(x, off, 32);
  return x;
}

// feats0 = relu(features) -> out[:, 0:128]
__global__ void k_relu0(const float* __restrict__ f, float* __restrict__ out, int n) {
  int idx = blockIdx.x * blockDim.x + threadIdx.x;
  if (idx >= n * DIM) return;
  int node = idx / DIM, d = idx - node * DIM;
  out[(long long)node * DIMO + d] = fmaxf(f[idx], 0.0f);
}

// zero next-layer columns, init segment max/sum
__global__ void k_prep(float* __restrict__ out, float* __restrict__ m,
                       float* __restrict__ s, int l) {
  int node = blockIdx.x, d = threadIdx.x;
  out[(long long)node * DIMO + (l + 1) * DIM + d] = 0.0f;
  if (d == 0) m[node] = -__builtin_inff();
  if (d == 1) s[node] = 0.0f;
}

// pass A: att[t] = dot(l2norm(sv*rel_emb[relid]), attn[l]); segment max into m[row]
__global__ void k_attA(const float* __restrict__ rel_emb,
                       const long long* __restrict__ spi,
                       const float* __restrict__ spv,
                       const float* __restrict__ attn,
                       const long long* __restrict__ adj,
                       float* __restrict__ att, float* __restrict__ m,
                       int l, int T) {
  int t    = (blockIdx.x * blockDim.x + threadIdx.x) >> 5;  // wave per triple
  int lane = threadIdx.x & 31;
  if (t >= T) return;
  int   rel = (int)spi[2 * t + 1];
  float sv  = spv[t];
  float4 r  = *(const float4*)(rel_emb + (long long)rel * DIM + lane * 4);
  r.x *= sv; r.y *= sv; r.z *= sv; r.w *= sv;
  float ss = waveRedSum(r.x * r.x + r.y * r.y + r.z * r.z + r.w * r.w);
  float rn = rsqrtf(fmaxf(ss, 1e-12f));
  float4 ak = *(const float4*)(attn + l * DIM + lane * 4);
  float av  = waveRedSum(r.x * ak.x + r.y * ak.y + r.z * ak.z + r.w * ak.w) * rn;
  if (lane == 0) {
    att[t] = av;
    int row = (int)adj[2 * t];
    atomicMaxF(m + row, av);
  }
}

// pass B: e = exp(att - m[row]); att <- e; s[row] += e
__global__ void k_expB(float* __restrict__ att, const float* __restrict__ m,
                       float* __restrict__ s, const long long* __restrict__ adj, int T) {
  int t = blockIdx.x * blockDim.x + threadIdx.x;
  if (t >= T) return;
  int   row = (int)adj[2 * t];
  float e   = expf(att[t] - m[row]);
  att[t]    = e;
  unsafeAtomicAdd(s + row, e);
}

// pass C: reflect neighbors, scale by softmax weight, scatter-add into next layer
__global__ void k_scatC(const float* __restrict__ rel_emb,
                        const long long* __restrict__ spi,
                        const float* __restrict__ spv,
                        const long long* __restrict__ adj,
                        const float* __restrict__ att,
                        const float* __restrict__ s,
                        float* __restrict__ out, int l, int T) {
  int t    = (blockIdx.x * blockDim.x + threadIdx.x) >> 5;
  int lane = threadIdx.x & 31;
  if (t >= T) return;
  int   rel = (int)spi[2 * t + 1];
  float sv  = spv[t];
  float4 r  = *(const float4*)(rel_emb + (long long)rel * DIM + lane * 4);
  r.x *= sv; r.y *= sv; r.z *= sv; r.w *= sv;
  float ss = waveRedSum(r.x * r.x + r.y * r.y + r.z * r.z + r.w * r.w);
  float rn = rsqrtf(fmaxf(ss, 1e-12f));
  r.x *= rn; r.y *= rn; r.z *= rn; r.w *= rn;
  int row = (int)adj[2 * t];
  int col = (int)adj[2 * t + 1];
  float4 nb = *(const float4*)(out + (long long)col * DIMO + l * DIM + lane * 4);
  float dt2 = 2.0f * waveRedSum(nb.x * r.x + nb.y * r.y + nb.z * r.z + nb.w * r.w);
  float a   = att[t] / s[row];
  float* dst = out + (long long)row * DIMO + (l + 1) * DIM + lane * 4;
  unsafeAtomicAdd(dst + 0, (nb.x - dt2 * r.x) * a);
  unsafeAtomicAdd(dst + 1, (nb.y - dt2 * r.y) * a);
  unsafeAtomicAdd(dst + 2, (nb.z - dt2 * r.z) * a);
  unsafeAtomicAdd(dst + 3, (nb.w - dt2 * r.w) * a);
}

// in-place relu on next-layer columns
__global__ void k_relu1(float* __restrict__ out, int l, int n) {
  int idx = blockIdx.x * blockDim.x + threadIdx.x;
  if (idx >= n * DIM) return;
  int node = idx / DIM, d = idx - node * DIM;
  float* p = out + (long long)node * DIMO + (l + 1) * DIM + d;
  *p = fmaxf(*p, 0.0f);
}

// row-wise l2 normalization of proxy -> npx (64 x 384)
__global__ void k_nproxy(const float* __restrict__ proxy, float* __restrict__ npx) {
  __shared__ float red[128];
  int row = blockIdx.x, tid = threadIdx.x;
  float p0 = proxy[row * DIMO + tid];
  float p1 = proxy[row * DIMO + tid + 128];
  float p2 = proxy[row * DIMO + tid + 256];
  red[tid] = p0 * p0 + p1 * p1 + p2 * p2;
  __syncthreads();
  for (int off = 64; off > 0; off >>= 1) {
    if (tid < off) red[tid] += red[tid + off];
    __syncthreads();
  }
  float rn = rsqrtf(fmaxf(red[0], 1e-12f));
  npx[row * DIMO + tid]       = p0 * rn;
  npx[row * DIMO + tid + 128] = p1 * rn;
  npx[row * DIMO + tid + 256] = p2 * rn;
}

// Fused head: logits -> softmax -> proxy_feature -> gate -> blend.
// Block = 128 threads (4 waves), 16 rows per block. f32 WMMA 16x16x4.
__global__ __launch_bounds__(128) void k_final(float* __restrict__ out,
                                               const float* __restrict__ proxy,
                                               const float* __restrict__ npx,
                                               const float* __restrict__ gk,
                                               int nn) {
  __shared__ float s_out[16][DIMO];
  __shared__ float s_pf[16][DIMO];
  __shared__ float s_att[16][NPROXY];
  __shared__ float s_rn[16];
  int tid = threadIdx.x;
  long long base = (long long)blockIdx.x * 16;

  for (int idx = tid; idx < 16 * DIMO; idx += 128) {
    int r = idx / DIMO, d = idx - r * DIMO;
    long long gr = base + r; if (gr >= nn) gr = nn - 1;
    s_out[r][d] = out[gr * DIMO + d];
  }
  if (tid < 16) s_rn[tid] = 0.0f;
  __syncthreads();
  {
    int r = tid >> 3, g = tid & 7;
    float p = 0.0f;
    for (int d = g * 48; d < g * 48 + 48; ++d) { float v = s_out[r][d]; p += v * v; }
    atomicAdd(&s_rn[r], p);
  }
  __syncthreads();
  if (tid < 16) s_rn[tid] = rsqrtf(fmaxf(s_rn[tid], 1e-12f));
  __syncthreads();

  int wave = tid >> 5, lane = tid & 31;
  int mr   = lane & 15;                 // row (A) or col (B/C) within tile
  int koff = (lane < 16) ? 0 : 2;       // K split across lane halves
  int mhi  = (lane < 16) ? 0 : 8;       // C/D row offset for upper lanes

  // ---- logits = l2norm(out) @ l2norm(proxy)^T  (16 x 64, K=384) ----
  {
    int n0 = wave * 16;
    int n  = n0 + mr;
    v8f c = {0.f, 0.f, 0.f, 0.f, 0.f, 0.f, 0.f, 0.f};
    float rnm = s_rn[mr];
    for (int k0 = 0; k0 < DIMO; k0 += 4) {
      v2f a, b;
      a.x = s_out[mr][k0 + koff] * rnm;
      a.y = s_out[mr][k0 + koff + 1] * rnm;
      b.x = npx[n * DIMO + k0 + koff];
      b.y = npx[n * DIMO + k0 + koff + 1];
      c = __builtin_amdgcn_wmma_f32_16x16x4_f32(false, a, false, b, (short)0, c,
                                                false, false);
    }
#pragma unroll
    for (int v = 0; v < 8; ++v) s_att[v + mhi][n0 + mr] = c[v];
  }
  __syncthreads();

  // ---- row softmax over 64 proxies ----
  if (tid < 16) {
    float mx = -__builtin_inff();
    for (int j = 0; j < NPROXY; ++j) mx = fmaxf(mx, s_att[tid][j]);
    float sm = 0.0f;
    for (int j = 0; j < NPROXY; ++j) { float e = expf(s_att[tid][j] - mx); s_att[tid][j] = e; sm += e; }
    float inv = 1.0f / sm;
    for (int j = 0; j < NPROXY; ++j) s_att[tid][j] *= inv;
  }
  __syncthreads();

  // ---- proxy_feature = out - proxy_att @ proxy  (16 x 384, K=64) ----
  for (int t = 0; t < 6; ++t) {
    int n0 = (wave + 4 * t) * 16;
    int n  = n0 + mr;
    v8f c = {0.f, 0.f, 0.f, 0.f, 0.f, 0.f, 0.f, 0.f};
    for (int k0 = 0; k0 < NPROXY; k0 += 4) {
      v2f a, b;
      a.x = s_att[mr][k0 + koff];
      a.y = s_att[mr][k0 + koff + 1];
      b.x = proxy[(k0 + koff) * DIMO + n];
      b.y = proxy[(k0 + koff + 1) * DIMO + n];
      c = __builtin_amdgcn_wmma_f32_16x16x4_f32(false, a, false, b, (short)0, c,
                                                false, false);
    }
#pragma unroll
    for (int v = 0; v < 8; ++v) {
      int M = v + mhi;
      s_pf[M][n] = s_out[M][n] - c[v];
    }
  }
  __syncthreads();

  // ---- gate = sigmoid(pf @ gate_kernel); result = g*out + (1-g)*pf ----
  for (int t = 0; t < 6; ++t) {
    int n0 = (wave + 4 * t) * 16;
    int n  = n0 + mr;
    v8f c = {0.f, 0.f, 0.f, 0.f, 0.f, 0.f, 0.f, 0.f};
    for (int k0 = 0; k0 < DIMO; k0 += 4) {
      v2f a, b;
      a.x = s_pf[mr][k0 + koff];
      a.y = s_pf[mr][k0 + koff + 1];
      b.x = gk[(k0 + koff) * DIMO + n];
      b.y = gk[(k0 + koff + 1) * DIMO + n];
      c = __builtin_amdgcn_wmma_f32_16x16x4_f32(false, a, false, b, (short)0, c,
                                                false, false);
    }
#pragma unroll
    for (int v = 0; v < 8; ++v) {
      int M = v + mhi;
      long long gr = base + M; if (gr >= nn) gr = nn - 1;
      float g = 1.0f / (1.0f + expf(-c[v]));
      float o = s_out[M][n], p = s_pf[M][n];
      out[gr * DIMO + n] = g * o + (1.0f - g) * p;
    }
  }
}

extern "C" void kernel_launch(void* const* d_in, const int* in_sizes, int n_in,
                              void* d_out, int out_size, void* d_ws, size_t ws_size,
                              hipStream_t stream) {
  const float*     features = (const float*)d_in[0];
  const float*     rel_emb  = (const float*)d_in[1];
  const long long* adj      = (const long long*)d_in[2];
  const long long* spi      = (const long long*)d_in[3];
  const float*     spv      = (const float*)d_in[4];
  const float*     attn     = (const float*)d_in[5];
  const float*     proxy    = (const float*)d_in[6];
  const float*     gk       = (const float*)d_in[7];
  float* out = (float*)d_out;

  int n = in_sizes[0] / DIM;  // 100000 nodes
  int T = in_sizes[4];        // 1000000 triples

  float* att = (float*)d_ws;  // T floats
  float* m   = att + T;       // n floats
  float* s   = m + n;         // n floats
  float* npx = s + n;         // 64*384 floats

  k_relu0<<<(n * DIM + 255) / 256, 256, 0, stream>>>(features, out, n);
  for (int l = 0; l < 2; ++l) {
    k_prep<<<n, DIM, 0, stream>>>(out, m, s, l);
    k_attA<<<(T + 7) / 8, 256, 0, stream>>>(rel_emb, spi, spv, attn, adj, att, m, l, T);
    k_expB<<<(T + 255) / 256, 256, 0, stream>>>(att, m, s, adj, T);
    k_scatC<<<(T + 7) / 8, 256, 0, stream>>>(rel_emb, spi, spv, adj, att, s, out, l, T);
    k_relu1<<<(n * DIM + 255) / 256, 256, 0, stream>>>(out, l, n);
  }
  k_nproxy<<<NPROXY, 128, 0, stream>>>(proxy, npx);
  k_final<<<(n + 15) / 16, 128, 0, stream>>>(out, proxy, npx, gk, n);
}